// SelfMixTP_29154238005853
// MI455X (gfx1250) — compile-verified
//
#include <hip/hip_runtime.h>

typedef __attribute__((ext_vector_type(2))) float v2f;
typedef __attribute__((ext_vector_type(4))) float v4f;
typedef __attribute__((ext_vector_type(8))) float v8f;

// ======================================================================
// Compile-time real Wigner 3j generation (e3nn convention), evaluated by
// the C++ frontend via constexpr; values constant-fold into the kernel.
// ======================================================================
namespace w3 {

constexpr double fact(int n){ double r=1.0; for(int i=2;i<=n;++i) r*=(double)i; return r; }

constexpr double csqrt(double x){
  if (x<=0.0) return 0.0;
  double r = x>=1.0 ? x : 1.0;
  double prev = 0.0;
  for (int i=0;i<100;++i){ if (r==prev) break; prev=r; r=0.5*(r+x/r); }
  return r;
}

constexpr double cabsd(double x){ return x<0.0?-x:x; }

constexpr double su2(int j1,int j2,int j3,int m1,int m2,int m3){
  if (m1+m2!=m3) return 0.0;
  int lo=j1-j2; if (lo<0) lo=-lo;
  if (j3<lo || j3>j1+j2) return 0.0;
  double pre = csqrt((2.0*j3+1.0)*fact(j3+j1-j2)*fact(j3-j1+j2)*fact(j1+j2-j3)/fact(j1+j2+j3+1));
  pre *= csqrt(fact(j3+m3)*fact(j3-m3)*fact(j1-m1)*fact(j1+m1)*fact(j2-m2)*fact(j2+m2));
  int kmin=0; if (j2-j3-m1>kmin) kmin=j2-j3-m1; if (j1-j3+m2>kmin) kmin=j1-j3+m2;
  int kmax=j1+j2-j3; if (j1-m1<kmax) kmax=j1-m1; if (j2+m2<kmax) kmax=j2+m2;
  double s=0.0;
  for (int k=kmin;k<=kmax;++k){
    double d = fact(k)*fact(j1+j2-j3-k)*fact(j1-m1-k)*fact(j2+m2-k)*fact(j3-j2+m1+k)*fact(j3-j1-m2+k);
    s += ((k&1)?-1.0:1.0)/d;
  }
  return pre*s;
}

struct CM { double re[9][9]; double im[9][9]; };

constexpr CM qmat(int l){
  CM q{};
  const double is2 = 1.0/csqrt(2.0);
  for (int m=-l;m<0;++m){ q.re[l+m][l-m]=is2; q.im[l+m][l+m]=-is2; }
  q.re[l][l]=1.0;
  for (int m=1;m<=l;++m){ double s=(m&1)?-1.0:1.0; q.re[l+m][l+m]=s*is2; q.im[l+m][l-m]=s*is2; }
  double fr=1.0, fi=0.0;               // (-1j)^l
  const int lm=l&3;
  if (lm==1){fr=0.0;fi=-1.0;} else if (lm==2){fr=-1.0;fi=0.0;} else if (lm==3){fr=0.0;fi=1.0;}
  CM o{};
  for (int i=0;i<2*l+1;++i) for (int j=0;j<2*l+1;++j){
    o.re[i][j]=fr*q.re[i][j]-fi*q.im[i][j];
    o.im[i][j]=fr*q.im[i][j]+fi*q.re[i][j];
  }
  return o;
}

struct W3 { float v[9][9][9]; };

constexpr W3 w3j(int l1,int l2,int l3){
  const int n1=2*l1+1, n2=2*l2+1, n3=2*l3+1;
  int nb[32]={}, nd[32]={}, nf[32]={}; double nv[32]={}; int nn=0;
  for (int m1=-l1;m1<=l1;++m1) for (int m2=-l2;m2<=l2;++m2){
    const int m3=m1+m2;
    if (m3>=-l3 && m3<=l3){
      const double c=su2(l1,l2,l3,m1,m2,m3);
      if (c!=0.0){ nb[nn]=l1+m1; nd[nn]=l2+m2; nf[nn]=l3+m3; nv[nn]=c; ++nn; }
    }
  }
  const CM q1=qmat(l1), q2=qmat(l2), q3=qmat(l3);
  double Rre[9][9][9]={}, Rim[9][9][9]={};
  double sre=0.0, sim=0.0;
  for (int a=0;a<n1;++a) for (int c=0;c<n2;++c) for (int e=0;e<n3;++e){
    double tr=0.0, ti=0.0;
    for (int t=0;t<nn;++t){
      const double r1=q1.re[a][nb[t]], i1=q1.im[a][nb[t]];
      const double r2=q2.re[c][nd[t]], i2=q2.im[c][nd[t]];
      const double r3=q3.re[e][nf[t]], i3=-q3.im[e][nf[t]];  // conj(q3)
      const double rr=r1*r2-i1*i2, ri=r1*i2+i1*r2;
      tr += (rr*r3-ri*i3)*nv[t];
      ti += (rr*i3+ri*r3)*nv[t];
    }
    Rre[a][c][e]=tr; Rim[a][c][e]=ti;
    sre+=cabsd(tr); sim+=cabsd(ti);
  }
  const bool useRe = (sre>=sim);
  double nrm2=0.0;
  for (int a=0;a<n1;++a) for (int c=0;c<n2;++c) for (int e=0;e<n3;++e){
    const double v = useRe?Rre[a][c][e]:Rim[a][c][e];
    nrm2 += v*v;
  }
  const double inv = 1.0/csqrt(nrm2);
  double best=-1.0, bestval=0.0;   // np.argmax: first occurrence of max |v|
  for (int a=0;a<n1;++a) for (int c=0;c<n2;++c) for (int e=0;e<n3;++e){
    const double v = useRe?Rre[a][c][e]:Rim[a][c][e];
    const double av=cabsd(v);
    if (av>best){best=av;bestval=v;}
  }
  const double sgn = bestval<0.0 ? -1.0 : 1.0;
  W3 o{};
  for (int a=0;a<n1;++a) for (int c=0;c<n2;++c) for (int e=0;e<n3;++e)
    o.v[a][c][e]=(float)((useRe?Rre[a][c][e]:Rim[a][c][e])*inv*sgn);
  return o;
}

// separate constexpr variables -> separate constant evaluations (step limit)
constexpr W3 W000=w3j(0,0,0);
constexpr W3 W022=w3j(0,2,2);
constexpr W3 W110=w3j(1,1,0);
constexpr W3 W112=w3j(1,1,2);
constexpr W3 W122=w3j(1,2,2);
constexpr W3 W123=w3j(1,2,3);
constexpr W3 W202=w3j(2,0,2);
constexpr W3 W212=w3j(2,1,2);
constexpr W3 W213=w3j(2,1,3);
constexpr W3 W220=w3j(2,2,0);
constexpr W3 W222=w3j(2,2,2);
constexpr W3 W223=w3j(2,2,3);
constexpr W3 W224=w3j(2,2,4);

struct Entry { unsigned char a,b,c,pad; float v; };
struct Table { int start[14]; Entry e[1024]; };

// INSTR order: (0,0,0e) (0,2,2e) (1,1,0e) (1,1,2e) (1,2,2o) (1,2,3o)
//              (2,0,2e) (2,1,2o) (2,1,3o) (2,2,0e) (2,2,2e) (2,2,3e) (2,2,4e)
constexpr int PL1[13]={0,0,1,1,1,1,2,2,2,2,2,2,2};
constexpr int PL2[13]={0,2,1,1,2,2,0,1,1,2,2,2,2};
constexpr int PL3[13]={0,2,0,2,2,3,2,2,3,0,2,3,4};
// key index: 0:(0,e) 1:(2,o) 2:(2,e) 3:(3,o) 4:(3,e) 5:(4,e)
constexpr int PKEY[13]={0,2,0,2,1,3,2,1,3,0,2,4,5};

constexpr Table make_table(){
  Table T{};
  const W3* ws[13]={&W000,&W022,&W110,&W112,&W122,&W123,&W202,&W212,&W213,&W220,&W222,&W223,&W224};
  const int kl[6]={0,2,2,3,3,4};
  const int kn[6]={3,2,4,2,1,1};      // path counts per output key
  int cnt=0;
  for (int p=0;p<13;++p){
    T.start[p]=cnt;
    const double alpha = csqrt((2.0*kl[PKEY[p]]+1.0)/(double)kn[PKEY[p]]);
    const int n1=2*PL1[p]+1, n2=2*PL2[p]+1, n3=2*PL3[p]+1;
    for (int a=0;a<n1;++a) for (int b=0;b<n2;++b) for (int c=0;c<n3;++c){
      const double v=(double)ws[p]->v[a][b][c];
      if (v>1e-6 || v<-1e-6){
        T.e[cnt].a=(unsigned char)a; T.e[cnt].b=(unsigned char)b;
        T.e[cnt].c=(unsigned char)c; T.e[cnt].pad=0;
        T.e[cnt].v=(float)(v*alpha);
        ++cnt;
      }
    }
  }
  T.start[13]=cnt;
  return T;
}

constexpr Table TPT = make_table();
} // namespace w3

// per-path hidden component bases ({0e:0, 1o:1..3, 2e:4..8})
constexpr int PIB[13]={0,0,1,1,1,1,4,4,4,4,4,4,4};
constexpr int PJB[13]={0,4,1,1,4,4,0,1,1,4,4,4,4};
// output keys: acc-offset, dim, global block base (u-major within key)
constexpr int KOB [6]={0,1,6,11,18,25};
constexpr int KDIM[6]={1,5,5,7,7,9};
constexpr int KGBLK[6]={0,32,192,352,576,800};

// ======================================================================
// Per-side linear: hidden = X @ W / sqrt(fan_in), v_wmma_f32_16x16x4_f32.
// Direct kernel-arg pointers so loads lower to global_load (not flat).
// k-outer / m-inner: B fragment loaded once per k-step, A loads for the
// m components are contiguous runs the compiler merges into wide loads.
// ======================================================================
__device__ __forceinline__ void gemm_side(
    const float* __restrict__ xr,
    const float* __restrict__ w0, const float* __restrict__ w1, const float* __restrict__ w2,
    float* __restrict__ Hs, const int side, const int ncol, const int hl)
{
  constexpr int MULS[3]={128,64,32};
  constexpr int CT  [3]={1,3,5};
  constexpr int BASE[3]={0,128,320};
  constexpr int CB  [3]={0,1,4};   // component base in H
  const float SC[3]={0.0883883476483184467f, 0.125f, 0.1767766952966368932f}; // 1/sqrt(fan_in)

  #pragma unroll
  for (int t=0;t<3;++t){
    const float* __restrict__ wt = (t==0) ? w0 : ((t==1) ? w1 : w2);
    const int ct=CT[t], base=BASE[t];
    v8f c[5];
    #pragma unroll
    for (int m=0;m<CT[t];++m) c[m] = (v8f){0.f,0.f,0.f,0.f,0.f,0.f,0.f,0.f};
    for (int k0=0;k0<MULS[t];k0+=4){
      const int ka = k0 + 2*hl;
      v2f b;
      // B frag (4x16 f32, per-lane column): lane -> N=lane%16
      b.x = wt[ ka   *32 + ncol];
      b.y = wt[(ka+1)*32 + ncol];
      #pragma unroll
      for (int m=0;m<CT[t];++m){
        v2f a;
        // A frag (16x4 f32): lanes 0-15 -> K={k0,k0+1}; lanes 16-31 -> K={k0+2,k0+3}
        a.x = xr[base +  ka   *ct + m];
        a.y = xr[base + (ka+1)*ct + m];
        c[m] = __builtin_amdgcn_wmma_f32_16x16x4_f32(false, a, false, b, (short)0, c[m], false, false);
      }
    }
    const float sc = SC[t];
    #pragma unroll
    for (int m=0;m<CT[t];++m){
      // C frag: lanes 0-15 v[r] -> M=r; lanes 16-31 v[r] -> M=8+r; N=lane%16
      #pragma unroll
      for (int r=0;r<8;++r){
        const int nl = r + 8*hl;
        Hs[((nl*2+side)*9 + CB[t]+m)*32 + ncol] = c[m][r]*sc;
      }
    }
  }
}

// ======================================================================
// Fused kernel: WMMA f32 linears -> LDS -> sparse CG TP -> LDS out tile
// -> coalesced non-temporal b128 streaming stores.
// block = 128 threads (4 waves), 16-node tile per block.
// ======================================================================
__global__ __launch_bounds__(128) void selfmixtp_kernel(
    const float* __restrict__ x,
    const float* __restrict__ wu0, const float* __restrict__ wu1, const float* __restrict__ wu2,
    const float* __restrict__ wv0, const float* __restrict__ wv1, const float* __restrict__ wv2,
    const float* __restrict__ tpw,
    float* __restrict__ out, int nodes)
{
  __shared__ __align__(16) float Hs[16*2*9*32];  // hidden [node][side][comp][u]   36.9 KB
  __shared__ __align__(16) float Os[16*1088];    // output tile [node][1088]       69.6 KB

  const int tid  = threadIdx.x;
  const int lane = tid & 31;
  const int wid  = tid >> 5;
  const int side = wid >> 1;        // 0 = U linear, 1 = V linear
  const int nt   = wid & 1;         // which 16-wide u-tile
  const int ln16 = lane & 15;
  const int hl   = lane >> 4;       // half-wave: K pair select (A/B frag layout)
  const int ncol = nt*16 + ln16;    // hidden u column this lane produces
  const int n0   = blockIdx.x * 16;

  int arow = n0 + ln16; if (arow > nodes-1) arow = nodes-1;  // clamp tail
  const float* __restrict__ xr = x + (size_t)arow * 480;

  // ---- phase 1: U / V linears via WMMA ----
  if (side==0) gemm_side(xr, wu0, wu1, wu2, Hs, 0, ncol, hl);
  else         gemm_side(xr, wv0, wv1, wv2, Hs, 1, ncol, hl);
  __syncthreads();

  // ---- phase 2: 13 CG tensor-product paths, one (node,u) pair per thread-iter ----
  const int u = tid & 31;
  float w13[13];
  #pragma unroll
  for (int p=0;p<13;++p) w13[p]=tpw[p*32+u];

  for (int it=0; it<4; ++it){
    const int nl = (tid>>5) + 4*it;
    float xu[9], xv[9];
    #pragma unroll
    for (int c2=0;c2<9;++c2){
      xu[c2]=Hs[((nl*2+0)*9+c2)*32+u];
      xv[c2]=Hs[((nl*2+1)*9+c2)*32+u];
    }
    float acc[34];
    #pragma unroll
    for (int o=0;o<34;++o) acc[o]=0.f;
    #pragma unroll
    for (int p=0;p<13;++p){
      float pacc[9];
      #pragma unroll
      for (int c2=0;c2<9;++c2) pacc[c2]=0.f;
      #pragma unroll
      for (int i=w3::TPT.start[p]; i<w3::TPT.start[p+1]; ++i){
        const int ea=w3::TPT.e[i].a, eb=w3::TPT.e[i].b, ec=w3::TPT.e[i].c;
        pacc[ec] += w3::TPT.e[i].v * xu[PIB[p]+ea] * xv[PJB[p]+eb];
      }
      const int k=w3::PKEY[p];
      #pragma unroll
      for (int c2=0;c2<KDIM[k];++c2) acc[KOB[k]+c2] += w13[p]*pacc[c2];
    }
    // stage this (node,u)'s 34 outputs in the LDS output tile
    float* __restrict__ orow = Os + nl*1088;
    #pragma unroll
    for (int k=0;k<6;++k){
      #pragma unroll
      for (int c2=0;c2<KDIM[k];++c2)
        orow[KGBLK[k] + u*KDIM[k] + c2] = acc[KOB[k]+c2];
    }
  }
  __syncthreads();

  // ---- phase 3: stream the contiguous 16x1088 row block to global ----
  // 16B-aligned, fully coalesced, non-temporal (write-once; keep L2 for x/weights)
  const size_t obase = (size_t)n0 * 1088;
  if (n0 + 16 <= nodes){
    for (int it=0; it<34; ++it){
      const int idx = it*512 + tid*4;           // 34*512 = 17408 = 16*1088
      const v4f val = *(const v4f*)&Os[idx];
      __builtin_nontemporal_store(val, (v4f*)(out + obase + idx));
    }
  } else {
    for (int it=0; it<34; ++it){
      const int idx = it*512 + tid*4;
      if (n0 + idx/1088 < nodes){
        const v4f val = *(const v4f*)&Os[idx];
        __builtin_nontemporal_store(val, (v4f*)(out + obase + idx));
      }
    }
  }
}

extern "C" void kernel_launch(void* const* d_in, const int* in_sizes, int n_in,
                              void* d_out, int out_size, void* d_ws, size_t ws_size,
                              hipStream_t stream) {
  (void)n_in; (void)out_size; (void)d_ws; (void)ws_size;
  const float* x   = (const float*)d_in[0];
  const float* wu0 = (const float*)d_in[1];
  const float* wu1 = (const float*)d_in[2];
  const float* wu2 = (const float*)d_in[3];
  const float* wv0 = (const float*)d_in[4];
  const float* wv1 = (const float*)d_in[5];
  const float* wv2 = (const float*)d_in[6];
  const float* tpw = (const float*)d_in[7];
  float* out = (float*)d_out;
  const int nodes = in_sizes[0] / 480;
  const int nblk  = (nodes + 15) / 16;
  selfmixtp_kernel<<<nblk, 128, 0, stream>>>(x, wu0,wu1,wu2, wv0,wv1,wv2, tpw, out, nodes);
}